// GINet_30717606101111
// MI455X (gfx1250) — compile-verified
//
#include <hip/hip_runtime.h>

// ---------------------------------------------------------------------------
// GIN forward for MI455X (gfx1250).  fp32 WMMA (V_WMMA_F32_16X16X4_F32) for
// all GEMMs; LDS-fused MLP (needs CDNA5's 320KB LDS); native f32 atomics for
// graph scatter.  N=100000, E=400000, G=4096, EMB=300, FEAT=512, L=5.
// ---------------------------------------------------------------------------

#define NN   100000
#define EENUM 400000
#define GG   4096
#define EMB  300
#define FEAT 512
#define NLAY 5
#define BN_EPS 1e-5f

// padded dims
#define C1P 608   // 600 -> 608 (38 tiles of 16)
#define C2P 304   // 300 -> 304 (19 tiles of 16)

typedef float v2f __attribute__((ext_vector_type(2)));
typedef float v8f __attribute__((ext_vector_type(8)));

// D = A(16x4) * B(4x16) + C ; fp32 WMMA, CDNA5 only.
// args: (neg_a, A, neg_b, B, c_mod, C, reuse_a, reuse_b)
__device__ __forceinline__ v8f wmma_f32(v2f a, v2f b, v8f c) {
  return __builtin_amdgcn_wmma_f32_16x16x4_f32(false, a, false, b, (short)0, c,
                                               false, false);
}

__device__ __forceinline__ void gAtomAdd(float* p, float v) {
  unsafeAtomicAdd(p, v);      // GLOBAL_ATOMIC_ADD_F32
}

// ---------------------------------------------------------------------------
// weight padding:  w1p [300][608], b1p[608];  w2p [608][304], b2p[304]
// ---------------------------------------------------------------------------
__global__ void pad_w1_kernel(const float* __restrict__ w1,
                              const float* __restrict__ b1,
                              float* __restrict__ w1p, float* __restrict__ b1p) {
  unsigned gid = blockIdx.x * 256u + threadIdx.x;
  const unsigned total = NLAY * EMB * C1P;
  if (gid >= total) return;
  unsigned l = gid / (EMB * C1P);
  unsigned rem = gid - l * (EMB * C1P);
  unsigned r = rem / C1P, c = rem - r * C1P;
  w1p[gid] = (c < 600u) ? w1[(l * EMB + r) * 600 + c] : 0.0f;
  if (gid < NLAY * C1P) {
    unsigned ll = gid / C1P, cc = gid - ll * C1P;
    b1p[gid] = (cc < 600u) ? b1[ll * 600 + cc] : 0.0f;
  }
}

__global__ void pad_w2_kernel(const float* __restrict__ w2,
                              const float* __restrict__ b2,
                              float* __restrict__ w2p, float* __restrict__ b2p) {
  unsigned gid = blockIdx.x * 256u + threadIdx.x;
  const unsigned total = NLAY * C1P * C2P;
  if (gid >= total) return;
  unsigned l = gid / (C1P * C2P);
  unsigned rem = gid - l * (C1P * C2P);
  unsigned k = rem / C2P, c = rem - k * C2P;
  w2p[gid] = (k < 600u && c < (unsigned)EMB) ? w2[(l * 600 + k) * EMB + c] : 0.0f;
  if (gid < NLAY * C2P) {
    unsigned ll = gid / C2P, cc = gid - ll * C2P;
    b2p[gid] = (cc < (unsigned)EMB) ? b2[ll * EMB + cc] : 0.0f;
  }
}

// ---------------------------------------------------------------------------
// node embedding: h[n][c] = atom_emb1[x[n,0]][c] + atom_emb2[x[n,1]][c]
// ---------------------------------------------------------------------------
__global__ void embed_kernel(const int* __restrict__ x,
                             const float* __restrict__ ae1,
                             const float* __restrict__ ae2,
                             float* __restrict__ H) {
  unsigned gid = blockIdx.x * 256u + threadIdx.x;
  if (gid >= (unsigned)NN * EMB) return;
  unsigned n = gid / EMB, c = gid - n * EMB;
  H[gid] = ae1[(unsigned)x[2 * n] * EMB + c] + ae2[(unsigned)x[2 * n + 1] * EMB + c];
}

// ---------------------------------------------------------------------------
// aggregation: agg[dst] += h[src] + edge_emb1[ea0] + edge_emb2[ea1]
// edges 0..E-1 real, E..E+N-1 self loops (attr = [4,0])
// ---------------------------------------------------------------------------
__global__ void aggregate_kernel(const float* __restrict__ H,
                                 float* __restrict__ Agg,
                                 const int* __restrict__ ei,
                                 const int* __restrict__ ea,
                                 const float* __restrict__ ee1,
                                 const float* __restrict__ ee2) {
  unsigned gid = blockIdx.x * 256u + threadIdx.x;
  const unsigned total = (unsigned)(EENUM + NN) * EMB;
  if (gid >= total) return;
  unsigned e = gid / EMB, c = gid - e * EMB;
  int src, dst, a0, a1;
  if (e < (unsigned)EENUM) {
    src = ei[e]; dst = ei[EENUM + e];
    a0 = ea[2 * e]; a1 = ea[2 * e + 1];
  } else {
    int nd = (int)(e - EENUM);
    src = nd; dst = nd; a0 = 4; a1 = 0;
  }
  float msg = H[(unsigned)src * EMB + c] + ee1[(unsigned)a0 * EMB + c]
            + ee2[(unsigned)a1 * EMB + c];
  gAtomAdd(&Agg[(unsigned)dst * EMB + c], msg);
}

// ---------------------------------------------------------------------------
// fused GIN MLP:  H = relu(A @ w1 + b1) @ w2 + b2  (per 64-row tile, in LDS)
// also accumulates per-column sum / sumsq for BatchNorm into stats[0..607],
// stats[608..1215].
// ---------------------------------------------------------------------------
__global__ __launch_bounds__(256) void fused_mlp_kernel(
    const float* __restrict__ A, float* __restrict__ H,
    const float* __restrict__ w1p, const float* __restrict__ b1p,
    const float* __restrict__ w2p, const float* __restrict__ b2p,
    float* __restrict__ stats) {
  extern __shared__ float smem[];
  float* As   = smem;                 // 64 x 304
  float* Zs   = As + 64 * C2P;        // 64 x 608
  float* cSum = Zs + 64 * C1P;        // 608
  float* cSq  = cSum + C1P;           // 608

  const int tid  = threadIdx.x;
  const int lane = tid & 31;
  const int wave = tid >> 5;
  const int lo   = lane & 15;
  const int hi   = lane >> 4;          // 0 or 1
  const int row0 = blockIdx.x * 64;

  for (int c = tid; c < 2 * C1P; c += 256) cSum[c] = 0.0f;

  // stage A tile (zero-pad rows >= NN, cols >= 300)
  for (int i = tid; i < 64 * C2P; i += 256) {
    int r = i / C2P, c = i - r * C2P;
    int gr = row0 + r;
    float v = 0.0f;
    if (c < EMB && gr < NN) v = A[(unsigned)gr * EMB + c];
    As[i] = v;
  }
  __syncthreads();

  // ---- stage 1: Z[64][608] = relu(A @ w1p + b1p) ----
  for (int t = wave; t < 4 * 38; t += 8) {
    int mt = t / 38, nt = t - mt * 38;
    int m0 = mt * 16, n0 = nt * 16;
    v8f acc = {0.f, 0.f, 0.f, 0.f, 0.f, 0.f, 0.f, 0.f};
    const float* Arow = As + (m0 + lo) * C2P;
    for (int k0 = 0; k0 < EMB; k0 += 4) {
      int ka = k0 + 2 * hi;
      v2f a = *(const v2f*)(Arow + ka);                  // (M=m0+lo, K=ka..ka+1)
      v2f b;
      b.x = w1p[ka * C1P + n0 + lo];                     // (K=ka,   N=n0+lo)
      b.y = w1p[(ka + 1) * C1P + n0 + lo];               // (K=ka+1, N=n0+lo)
      acc = wmma_f32(a, b, acc);
    }
    float bias = b1p[n0 + lo];
#pragma unroll
    for (int v = 0; v < 8; ++v) {
      float z = acc[v] + bias;
      Zs[(m0 + v + 8 * hi) * C1P + n0 + lo] = z > 0.0f ? z : 0.0f;
    }
  }
  __syncthreads();

  // ---- stage 2: Hout[64][300] = Z @ w2p + b2p  (+ BN partial stats) ----
  for (int t = wave; t < 4 * 19; t += 8) {
    int mt = t / 19, nt = t - mt * 19;
    int m0 = mt * 16, n0 = nt * 16;
    v8f acc = {0.f, 0.f, 0.f, 0.f, 0.f, 0.f, 0.f, 0.f};
    const float* Zrow = Zs + (m0 + lo) * C1P;
    for (int k0 = 0; k0 < C1P; k0 += 4) {
      int ka = k0 + 2 * hi;
      v2f a = *(const v2f*)(Zrow + ka);
      v2f b;
      b.x = w2p[ka * C2P + n0 + lo];
      b.y = w2p[(ka + 1) * C2P + n0 + lo];
      acc = wmma_f32(a, b, acc);
    }
    int n = n0 + lo;                 // <= 303, safe for b2p (304 entries)
    float bias = b2p[n];
    float ps = 0.0f, pq = 0.0f;
#pragma unroll
    for (int v = 0; v < 8; ++v) {
      int gr = row0 + m0 + v + 8 * hi;
      float hv = acc[v] + bias;
      if (gr < NN && n < EMB) {
        H[(unsigned)gr * EMB + n] = hv;
        ps += hv;
        pq += hv * hv;
      }
    }
    atomicAdd(&cSum[n], ps);         // DS float atomics
    atomicAdd(&cSq[n], pq);
  }
  __syncthreads();

  for (int c = tid; c < EMB; c += 256) {
    gAtomAdd(&stats[c], cSum[c]);
    gAtomAdd(&stats[C1P + c], cSq[c]);
  }
}

// ---------------------------------------------------------------------------
// BatchNorm finalize + apply
// ---------------------------------------------------------------------------
__global__ void bn_stats_kernel(const float* __restrict__ stats,
                                float* __restrict__ mu, float* __restrict__ inv) {
  int c = threadIdx.x;
  if (c < EMB) {
    float m = stats[c] * (1.0f / (float)NN);
    float var = stats[C1P + c] * (1.0f / (float)NN) - m * m;
    mu[c] = m;
    inv[c] = rsqrtf(var + BN_EPS);
  }
}

__global__ void bn_apply_kernel(float* __restrict__ H,
                                const float* __restrict__ gam,
                                const float* __restrict__ bet,
                                const float* __restrict__ mu,
                                const float* __restrict__ inv, int do_relu) {
  unsigned gid = blockIdx.x * 256u + threadIdx.x;
  if (gid >= (unsigned)NN * EMB) return;
  unsigned c = gid % (unsigned)EMB;
  float v = gam[c] * (H[gid] - mu[c]) * inv[c] + bet[c];
  if (do_relu) v = v > 0.0f ? v : 0.0f;
  H[gid] = v;
}

// ---------------------------------------------------------------------------
// global mean pool
// ---------------------------------------------------------------------------
__global__ void pool_accum_kernel(const float* __restrict__ H,
                                  const int* __restrict__ batch,
                                  float* __restrict__ pooled) {
  unsigned gid = blockIdx.x * 256u + threadIdx.x;
  if (gid >= (unsigned)NN * EMB) return;
  unsigned n = gid / EMB, c = gid - n * EMB;
  gAtomAdd(&pooled[(unsigned)batch[n] * EMB + c], H[gid]);
}

__global__ void pool_cnt_kernel(const int* __restrict__ batch,
                                float* __restrict__ cnts) {
  unsigned n = blockIdx.x * 256u + threadIdx.x;
  if (n < (unsigned)NN) gAtomAdd(&cnts[batch[n]], 1.0f);
}

__global__ void pool_div_kernel(float* __restrict__ pooled,
                                const float* __restrict__ cnts) {
  unsigned gid = blockIdx.x * 256u + threadIdx.x;
  if (gid >= (unsigned)GG * EMB) return;
  unsigned g = gid / EMB;
  pooled[gid] /= fmaxf(cnts[g], 1.0f);
}

// ---------------------------------------------------------------------------
// head GEMM: out[4096][512] = pooled[4096][300] @ feat_w[300][512] + feat_b
// one WMMA 16x16 tile per wave
// ---------------------------------------------------------------------------
__global__ __launch_bounds__(256) void final_gemm_kernel(
    const float* __restrict__ P, const float* __restrict__ W,
    const float* __restrict__ bias, float* __restrict__ out) {
  const int tid  = threadIdx.x;
  const int lane = tid & 31;
  const int wave = tid >> 5;
  const int lo   = lane & 15;
  const int hi   = lane >> 4;
  int tile = blockIdx.x * 8 + wave;            // 8192 tiles total
  if (tile >= (GG / 16) * (FEAT / 16)) return; // wave-uniform
  int mt = tile >> 5, nt = tile & 31;          // 32 col-tiles
  int m0 = mt * 16, n0 = nt * 16;
  v8f acc = {0.f, 0.f, 0.f, 0.f, 0.f, 0.f, 0.f, 0.f};
  const float* Prow = P + (unsigned)(m0 + lo) * EMB;
  for (int k0 = 0; k0 < EMB; k0 += 4) {
    int ka = k0 + 2 * hi;
    v2f a = *(const v2f*)(Prow + ka);
    v2f b;
    b.x = W[ka * FEAT + n0 + lo];
    b.y = W[(ka + 1) * FEAT + n0 + lo];
    acc = wmma_f32(a, b, acc);
  }
  float bb = bias[n0 + lo];
#pragma unroll
  for (int v = 0; v < 8; ++v)
    out[(unsigned)(m0 + v + 8 * hi) * FEAT + n0 + lo] = acc[v] + bb;
}

// ---------------------------------------------------------------------------
static inline int cdiv(long long a, long long b) { return (int)((a + b - 1) / b); }

extern "C" void kernel_launch(void* const* d_in, const int* in_sizes, int n_in,
                              void* d_out, int out_size, void* d_ws, size_t ws_size,
                              hipStream_t stream) {
  const int*   x     = (const int*)d_in[0];
  const int*   ei    = (const int*)d_in[1];
  const int*   ea    = (const int*)d_in[2];
  const int*   batch = (const int*)d_in[3];
  const float* ae1   = (const float*)d_in[4];
  const float* ae2   = (const float*)d_in[5];
  const float* ee1   = (const float*)d_in[6];
  const float* ee2   = (const float*)d_in[7];
  const float* w1    = (const float*)d_in[8];
  const float* b1    = (const float*)d_in[9];
  const float* w2    = (const float*)d_in[10];
  const float* b2    = (const float*)d_in[11];
  const float* gam   = (const float*)d_in[12];
  const float* bet   = (const float*)d_in[13];
  const float* fw    = (const float*)d_in[14];
  const float* fb    = (const float*)d_in[15];
  float* out = (float*)d_out;

  // workspace carve-up (floats)
  float* ws     = (float*)d_ws;
  float* bufH   = ws;                                  // 30,000,000
  float* bufA   = bufH + (size_t)NN * EMB;             // 30,000,000
  float* w1p    = bufA + (size_t)NN * EMB;             // 5*300*608
  float* b1p    = w1p + (size_t)NLAY * EMB * C1P;      // 5*608
  float* w2p    = b1p + (size_t)NLAY * C1P;            // 5*608*304
  float* b2p    = w2p + (size_t)NLAY * C1P * C2P;      // 5*304
  float* stats  = b2p + (size_t)NLAY * C2P;            // 1216
  float* bnmu   = stats + 2 * C1P;                     // 304
  float* bninv  = bnmu + C2P;                          // 304
  float* pooled = bninv + C2P;                         // 4096*300
  float* cnts   = pooled + (size_t)GG * EMB;           // 4096

  pad_w1_kernel<<<cdiv((long long)NLAY * EMB * C1P, 256), 256, 0, stream>>>(w1, b1, w1p, b1p);
  pad_w2_kernel<<<cdiv((long long)NLAY * C1P * C2P, 256), 256, 0, stream>>>(w2, b2, w2p, b2p);
  embed_kernel<<<cdiv((long long)NN * EMB, 256), 256, 0, stream>>>(x, ae1, ae2, bufH);

  const size_t ldsBytes = (size_t)(64 * C2P + 64 * C1P + 2 * C1P) * sizeof(float); // ~238 KB

  for (int l = 0; l < NLAY; ++l) {
    hipMemsetAsync(bufA, 0, sizeof(float) * (size_t)NN * EMB, stream);
    hipMemsetAsync(stats, 0, sizeof(float) * 2 * C1P, stream);
    aggregate_kernel<<<cdiv((long long)(EENUM + NN) * EMB, 256), 256, 0, stream>>>(
        bufH, bufA, ei, ea, ee1 + (size_t)l * 5 * EMB, ee2 + (size_t)l * 3 * EMB);
    fused_mlp_kernel<<<cdiv(NN, 64), 256, ldsBytes, stream>>>(
        bufA, bufH, w1p + (size_t)l * EMB * C1P, b1p + (size_t)l * C1P,
        w2p + (size_t)l * C1P * C2P, b2p + (size_t)l * C2P, stats);
    bn_stats_kernel<<<1, 512, 0, stream>>>(stats, bnmu, bninv);
    bn_apply_kernel<<<cdiv((long long)NN * EMB, 256), 256, 0, stream>>>(
        bufH, gam + (size_t)l * EMB, bet + (size_t)l * EMB, bnmu, bninv, (l < NLAY - 1) ? 1 : 0);
  }

  hipMemsetAsync(pooled, 0, sizeof(float) * (size_t)GG * EMB, stream);
  hipMemsetAsync(cnts, 0, sizeof(float) * GG, stream);
  pool_accum_kernel<<<cdiv((long long)NN * EMB, 256), 256, 0, stream>>>(bufH, batch, pooled);
  pool_cnt_kernel<<<cdiv(NN, 256), 256, 0, stream>>>(batch, cnts);
  pool_div_kernel<<<cdiv((long long)GG * EMB, 256), 256, 0, stream>>>(pooled, cnts);
  final_gemm_kernel<<<(GG / 16) * (FEAT / 16) / 8, 256, 0, stream>>>(pooled, fw, fb, out);
}